// Compute1AngleInput_81827716923457
// MI455X (gfx1250) — compile-verified
//
#include <hip/hip_runtime.h>

#define NN    8
#define NF    64
#define NPAIR 56                 // 8*8 - 8 off-diagonal pairs
#define NCOMP 195                // 3 + 3*64
#define CHUNK (NPAIR * NCOMP)    // 10920 floats per center (divisible by 4)

// Unified LDS pool layout (float indices):
//   [0, 64)        emb_i
//   [64, 576)      emb_j * (1/d)   (row j at 64 + j*64)
//   [576, 744)     per-pair header: {d_ij, d_ik, d_jk_norm} * 56 pairs
#define POOL_EMBJ  NF
#define POOL_HDR   (NF + NN * NF)
#define POOL_SIZE  (POOL_HDR + 3 * NPAIR)

typedef float v4f __attribute__((ext_vector_type(4)));

// One per-lane async global->LDS dword copy (CDNA5 GLOBAL_LOAD_ASYNC_TO_LDS_B32).
// lds_off: 32-bit LDS byte offset (low 32 bits of the generic shared pointer),
// gaddr:   64-bit global address (VGPR pair). Tracked by ASYNCcnt.
__device__ __forceinline__ void async_g2lds_b32(unsigned lds_off, const void* g) {
    asm volatile("global_load_async_to_lds_b32 %0, %1, off"
                 :: "v"(lds_off), "v"((unsigned long long)g)
                 : "memory");
}
__device__ __forceinline__ unsigned lds_off32(const void* p) {
    return (unsigned)(unsigned long long)p;   // generic LDS addr low 32 bits == LDS offset
}
__device__ __forceinline__ void wait_asynccnt0() {
#if __has_builtin(__builtin_amdgcn_s_wait_asynccnt)
    __builtin_amdgcn_s_wait_asynccnt(0);
#else
    asm volatile("s_wait_asynccnt 0" ::: "memory");
#endif
}

__global__ __launch_bounds__(256) void angle_feat_kernel(
    const int*   __restrict__ atom_i_idx,   // (nC)
    const int*   __restrict__ atom_j_idx,   // (nC*NN)
    const float* __restrict__ dist_ij,      // (nC*NN)
    const float* __restrict__ atoms_xyz,    // (N_ATOMS,3)
    const int*   __restrict__ atoms_long,   // (N_ATOMS,2)
    const float* __restrict__ embed_table,  // (16,NF)
    int*         __restrict__ out_idx,      // (nC) int view of d_out
    float*       __restrict__ out_ang)      // (nC,NPAIR,NCOMP)
{
    __shared__ float s_pool[POOL_SIZE];
    __shared__ float s_d[NN];
    __shared__ float s_invd[NN];
    __shared__ float s_xyz[NN][3];
    __shared__ int   s_spec[NN];
    __shared__ int   s_speci;

    const int c = blockIdx.x;
    const int t = threadIdx.x;

    // ---- Phase 1: per-center scalar gathers -------------------------------
    if (t < NN) {
        int   tj = atom_j_idx[c * NN + t];
        float d  = dist_ij[c * NN + t];
        s_d[t]    = d;
        s_invd[t] = 1.0f / d;
        s_spec[t] = atoms_long[tj * 2 + 1];
        s_xyz[t][0] = atoms_xyz[tj * 3 + 0];
        s_xyz[t][1] = atoms_xyz[tj * 3 + 1];
        s_xyz[t][2] = atoms_xyz[tj * 3 + 2];
    } else if (t == NN) {
        int ti = atom_i_idx[c];
        out_idx[c] = ti;
        s_speci = atoms_long[ti * 2 + 1];
    }
    __syncthreads();

    // ---- Phase 2: async embedding-row gathers straight into LDS -----------
    if (t < NF) {
        async_g2lds_b32(lds_off32(&s_pool[t]),
                        embed_table + (size_t)s_speci * NF + t);
    }
    #pragma unroll
    for (int e = t; e < NN * NF; e += 256) {
        async_g2lds_b32(lds_off32(&s_pool[POOL_EMBJ + e]),
                        embed_table + (size_t)s_spec[e >> 6] * NF + (e & (NF - 1)));
    }

    // Overlap with the async copies: per-pair header {d_ij, d_ik, d_jk_norm}.
    if (t < 3 * NPAIR) {
        int p  = t / 3;
        int c3 = t - p * 3;
        int j  = p / (NN - 1);
        int k0 = p - j * (NN - 1);
        int k  = k0 + (k0 >= j ? 1 : 0);
        float dx = s_xyz[j][0] - s_xyz[k][0];
        float dy = s_xyz[j][1] - s_xyz[k][1];
        float dz = s_xyz[j][2] - s_xyz[k][2];
        float djk  = sqrtf(dx * dx + dy * dy + dz * dz);
        float dj   = s_d[j], dk = s_d[k];
        float maxD = fmaxf(dj, dk);
        float minD = fminf(dj, dk);
        float norm = (djk - maxD + minD) / (2.0f * minD);
        s_pool[POOL_HDR + t] = (c3 == 0) ? dj : (c3 == 1) ? dk : norm;
    }

    wait_asynccnt0();
    __syncthreads();

    // Pre-scale emb_j rows by 1/d in place (both ej and ek use scaled rows).
    #pragma unroll
    for (int e = t; e < NN * NF; e += 256) {
        s_pool[POOL_EMBJ + e] *= s_invd[e >> 6];
    }
    __syncthreads();

    // ---- Phase 3: stream the 10920-float chunk as NT b128 stores ----------
    // Branchless decode: every element is exactly one ds_load_b32 from s_pool.
    float* chunk = out_ang + (size_t)c * CHUNK;   // 16B-aligned (CHUNK%4==0, base%4==0)
    for (int v = t; v < CHUNK / 4; v += 256) {
        v4f r;
        #pragma unroll
        for (int i = 0; i < 4; ++i) {
            int idx  = v * 4 + i;
            int p    = idx / NCOMP;               // magic-number division
            int comp = idx - p * NCOMP;
            int j    = p / (NN - 1);
            int k0   = p - j * (NN - 1);
            int k    = k0 + (k0 >= j ? 1 : 0);
            // embedding regions: [3,67)=emb_i, [67,131)=ej, [131,195)=ek
            bool r2   = comp >= 3 + NF;           // ej or ek
            bool r3   = comp >= 3 + 2 * NF;       // ek
            int  fo   = comp - (r3 ? (3 + 2 * NF) : (r2 ? (3 + NF) : 3));
            int  row  = r3 ? k : j;
            int  oemb = r2 ? (POOL_EMBJ + row * NF + fo) : fo;
            int  ohdr = POOL_HDR + p * 3 + comp;  // valid when comp < 3
            int  off  = (comp < 3) ? ohdr : oemb;
            r[i] = s_pool[off];
        }
        __builtin_nontemporal_store(r, (v4f*)(chunk + (size_t)v * 4));
    }
}

extern "C" void kernel_launch(void* const* d_in, const int* in_sizes, int n_in,
                              void* d_out, int out_size, void* d_ws, size_t ws_size,
                              hipStream_t stream) {
    // setup_inputs order:
    // 0: nNeigh (scalar, ==8, hardcoded), 1: atom_i_idx, 2: atom_j_idx,
    // 3: dist_ij, 4: atoms_xyz, 5: atoms_long, 6: embed_table
    const int*   atom_i_idx  = (const int*)  d_in[1];
    const int*   atom_j_idx  = (const int*)  d_in[2];
    const float* dist_ij     = (const float*)d_in[3];
    const float* atoms_xyz   = (const float*)d_in[4];
    const int*   atoms_long  = (const int*)  d_in[5];
    const float* embed_table = (const float*)d_in[6];

    const int nC = in_sizes[1];               // 10000

    int*   out_idx = (int*)d_out;             // first nC: atom_i_idx (int bits)
    float* out_ang = (float*)d_out + nC;      // then (nC, 56, 195) floats

    hipLaunchKernelGGL(angle_feat_kernel, dim3(nC), dim3(256), 0, stream,
                       atom_i_idx, atom_j_idx, dist_ij, atoms_xyz, atoms_long,
                       embed_table, out_idx, out_ang);
}